// AnisUpsampling_58935541236472
// MI455X (gfx1250) — compile-verified
//
#include <hip/hip_runtime.h>

// ---------------------------------------------------------------------------
// AnisUpsampling for MI455X (gfx1250).
//
// out0 (8,64,128,256) = ( upH(x0) + upW(x1) ) / 2
// out1 (8,64,256,128) = ( upH(x1) + upW(x2) ) / 2
//
// Tent 2x upsample (L=5, s=2) along one axis (m = j>>1, edges clamped):
//   j even: (x[m-1]+x[m]+x[m+1])/cnt
//   j odd : (x[m]+x[m+1])/cnt
//
// Memory-bound (~235 MB -> ~10us at 23.3 TB/s). This version targets the
// VALU-issue side so it stays under the HBM floor:
//   * each thread produces a 2(h) x 4(w) tile: the even/odd H row-pair shares
//     A rows q-1,q,q+1 -> 3 float4 loads yield 8 outputs
//   * divisors are only {1,2,3} -> multiply by selected reciprocal
//     (1.0/0.5 exact; 1/3 within 1 ulp of true divide)
//   * W-upsample source rows staged in LDS via gfx1250 async global->LDS
//     (one b128 per thread, ASYNCcnt + s_wait_asynccnt)
//   * global_prefetch_b8 warms L2 for the next A rows
// ---------------------------------------------------------------------------

#if __has_builtin(__builtin_amdgcn_global_load_async_to_lds_b128)
#define USE_ASYNC_128 1
typedef int v4i_t __attribute__((vector_size(4 * sizeof(int))));
typedef __attribute__((address_space(1))) v4i_t g_v4i;
typedef __attribute__((address_space(3))) v4i_t l_v4i;
#elif __has_builtin(__builtin_amdgcn_global_load_async_to_lds_b64)
#define USE_ASYNC_64 1
typedef int v2i_t __attribute__((vector_size(2 * sizeof(int))));
typedef __attribute__((address_space(1))) v2i_t g_v2i;
typedef __attribute__((address_space(3))) v2i_t l_v2i;
#endif

__device__ __forceinline__ void wait_async_then_barrier() {
#if defined(USE_ASYNC_128) || defined(USE_ASYNC_64)
#if __has_builtin(__builtin_amdgcn_s_wait_asynccnt)
    __builtin_amdgcn_s_wait_asynccnt(0);
#else
    asm volatile("s_wait_asynccnt 0x0" ::: "memory");
#endif
#endif
    __syncthreads();
}

template <int HOUT, int WOUT>
__global__ __launch_bounds__(256) void up_merge_kernel(
    const float* __restrict__ A,   // (BC, HOUT/2, WOUT)  -> upsampled along H
    const float* __restrict__ Bm,  // (BC, HOUT, WOUT/2)  -> upsampled along W
    float* __restrict__ out)       // (BC, HOUT, WOUT)
{
    constexpr int HIN = HOUT / 2;
    constexpr int WIN = WOUT / 2;
    constexpr int TPR = WOUT / 4;     // threads per output row-pair
    constexpr int PAIRS = 256 / TPR;  // h row-pairs per block
    constexpr int ROWS = PAIRS * 2;   // output rows per block
    static_assert(ROWS * WIN == 1024, "LDS sizing invariant");

    __shared__ float sB[1024];        // ROWS x WIN floats of Bm (4 KB)

    const int t = threadIdx.x;
    const int blocksPerImage = HOUT / ROWS;
    const int p = blockIdx.x / blocksPerImage;               // b*C + c plane
    const int hbase = (blockIdx.x % blocksPerImage) * ROWS;  // first out row

    // ---- stage Bm rows [hbase, hbase+ROWS) into LDS: 1024 floats, 4/thread
    {
        const int e = t * 4;                  // 16B-aligned, within one row
        const int srow = e / WIN;
        const int scol = e - srow * WIN;
        const float* gsrc =
            Bm + ((size_t)(p * HOUT + hbase + srow) * WIN + scol);
#if defined(USE_ASYNC_128)
        __builtin_amdgcn_global_load_async_to_lds_b128(
            (g_v4i*)const_cast<float*>(gsrc), (l_v4i*)&sB[e], 0, 0);
#elif defined(USE_ASYNC_64)
        __builtin_amdgcn_global_load_async_to_lds_b64(
            (g_v2i*)const_cast<float*>(gsrc), (l_v2i*)&sB[e], 0, 0);
        __builtin_amdgcn_global_load_async_to_lds_b64(
            (g_v2i*)const_cast<float*>(gsrc + 2), (l_v2i*)&sB[e + 2], 0, 0);
#else
        *(float4*)&sB[e] = *(const float4*)gsrc;
#endif
    }
    wait_async_then_barrier();

    // ---- this thread's 2x4 output tile: rows h0,h0+1, cols w0..w0+3
    const int pair = t / TPR;
    const int w0 = (t - pair * TPR) * 4;
    const int h0 = hbase + pair * 2;          // even
    const int q = h0 >> 1;                    // center source row in A

    // upH: even row h0 uses A[q-1],A[q],A[q+1]; odd row h0+1 uses A[q],A[q+1]
    const float* Aq = A + ((size_t)p * HIN + q) * WOUT + w0;
    const float4 aq = *(const float4*)Aq;
    float4 sE = aq, sO = aq;
    float cE = 1.0f, cO = 1.0f;
    if (q > 0) {
        const float4 am = *(const float4*)(Aq - WOUT);
        sE.x += am.x; sE.y += am.y; sE.z += am.z; sE.w += am.w;
        cE += 1.0f;
    }
    if (q + 1 < HIN) {
        const float4 ap = *(const float4*)(Aq + WOUT);
        sE.x += ap.x; sE.y += ap.y; sE.z += ap.z; sE.w += ap.w;
        sO.x += ap.x; sO.y += ap.y; sO.z += ap.z; sO.w += ap.w;
        cE += 1.0f; cO += 1.0f;
    }
    if (q + 2 < HIN) __builtin_prefetch(Aq + 2 * WOUT, 0, 0);

    const float invE = (cE == 3.0f) ? (1.0f / 3.0f) : 0.5f;  // cE in {2,3}
    const float invO = (cO == 2.0f) ? 0.5f : 1.0f;           // cO in {1,2}

    // upW from the LDS-staged Bm rows (same column window for both rows)
    const int m0 = w0 >> 1;
    const bool hasBm1 = (m0 > 0);
    const bool hasB2 = (m0 + 2 < WIN);
    const float r0 = hasBm1 ? (1.0f / 3.0f) : 0.5f;
    const float r2 = hasB2 ? (1.0f / 3.0f) : 0.5f;
    const float r3 = hasB2 ? 0.5f : 1.0f;

    const float* sRowE = &sB[(pair * 2) * WIN];
    float4 vW[2];
#pragma unroll
    for (int i = 0; i < 2; ++i) {
        const float* sRow = sRowE + i * WIN;
        const float bm1 = hasBm1 ? sRow[m0 - 1] : 0.0f;
        const float b0 = sRow[m0];
        const float b1 = sRow[m0 + 1];       // m0+1 <= WIN-1 always
        const float b2 = hasB2 ? sRow[m0 + 2] : 0.0f;
        vW[i].x = (bm1 + b0 + b1) * r0;      // w0   (even)
        vW[i].y = (b0 + b1) * 0.5f;          // w0+1 (odd)
        vW[i].z = (b0 + b1 + b2) * r2;       // w0+2 (even)
        vW[i].w = (b1 + b2) * r3;            // w0+3 (odd)
    }

    float* dst = out + ((size_t)p * HOUT + h0) * WOUT + w0;
    float4 oE, oO;
    oE.x = (sE.x * invE + vW[0].x) * 0.5f;
    oE.y = (sE.y * invE + vW[0].y) * 0.5f;
    oE.z = (sE.z * invE + vW[0].z) * 0.5f;
    oE.w = (sE.w * invE + vW[0].w) * 0.5f;
    oO.x = (sO.x * invO + vW[1].x) * 0.5f;
    oO.y = (sO.y * invO + vW[1].y) * 0.5f;
    oO.z = (sO.z * invO + vW[1].z) * 0.5f;
    oO.w = (sO.w * invO + vW[1].w) * 0.5f;
    *(float4*)dst = oE;
    *(float4*)(dst + WOUT) = oO;
}

extern "C" void kernel_launch(void* const* d_in, const int* in_sizes, int n_in,
                              void* d_out, int out_size, void* d_ws,
                              size_t ws_size, hipStream_t stream) {
    const float* x0 = (const float*)d_in[0];  // (8,64, 64,256)
    const float* x1 = (const float*)d_in[1];  // (8,64,128,128)
    const float* x2 = (const float*)d_in[2];  // (8,64,256, 64)

    constexpr int BC = 8 * 64;
    float* out0 = (float*)d_out;                              // (BC,128,256)
    float* out1 = out0 + (size_t)BC * 128 * 256;              // (BC,256,128)

    {   // out0 = (upH(x0) + upW(x1)) / 2
        constexpr int HOUT = 128, WOUT = 256;
        constexpr int ROWS = 2 * (256 / (WOUT / 4));
        dim3 grid(BC * HOUT / ROWS);                           // 8192 blocks
        up_merge_kernel<HOUT, WOUT><<<grid, 256, 0, stream>>>(x0, x1, out0);
    }
    {   // out1 = (upH(x1) + upW(x2)) / 2
        constexpr int HOUT = 256, WOUT = 128;
        constexpr int ROWS = 2 * (256 / (WOUT / 4));
        dim3 grid(BC * HOUT / ROWS);                           // 8192 blocks
        up_merge_kernel<HOUT, WOUT><<<grid, 256, 0, stream>>>(x1, x2, out1);
    }
}